// ConceptNet_65223373357442
// MI455X (gfx1250) — compile-verified
//
#include <hip/hip_runtime.h>

#define D_DIM 768
#define N_TRAIN 100000
#define NCON 64
#define BATCH 8192
#define CCLS 10

// output offsets (floats)
#define OFF_ORIG 0
#define OFF_YPRED (BATCH*CCLS)          // 81920
#define OFF_L1    (2*BATCH*CCLS)        // 163840
#define OFF_L2    (OFF_L1+1)
#define OFF_NM    (OFF_L1+2)
#define OFF_SP    (OFF_L1+3)

typedef __attribute__((ext_vector_type(2))) float v2f;
typedef __attribute__((ext_vector_type(8))) float v8f;

// ---------------------------------------------------------------------------
// Kernel 1: gram = C^T C, scalars, Gauss-Jordan inverse, Vmat = G^-1 (C^T W^T)
// One 1024-thread workgroup, everything in LDS.
// ---------------------------------------------------------------------------
__global__ __launch_bounds__(1024) void k_setup(const float* __restrict__ Cc,
                                                const float* __restrict__ W,
                                                float* __restrict__ out,
                                                float* __restrict__ ws) {
  __shared__ float aug[64][130];   // [gram | identity] augmented
  __shared__ float sU[64][CCLS];
  __shared__ float colp[64];
  __shared__ float red_sum, red_trace, red_abs;
  int tid = threadIdx.x;
  if (tid == 0) { red_sum = 0.f; red_trace = 0.f; red_abs = 0.f; }
  __syncthreads();

  float psum = 0.f, ptrc = 0.f, pabs = 0.f;
  for (int e = tid; e < 64*64; e += 1024) {
    int r = e >> 6, c = e & 63;
    float acc = 0.f;
    for (int d = 0; d < D_DIM; ++d) acc += Cc[d*NCON + r] * Cc[d*NCON + c];
    aug[r][c] = acc;
    aug[r][64 + c] = (r == c) ? 1.f : 0.f;
    psum += acc;
    if (r == c) ptrc += acc;
    pabs += fabsf(acc - ((r == c) ? 1.f : 0.f));
  }
  atomicAdd(&red_sum, psum);
  atomicAdd(&red_trace, ptrc);
  atomicAdd(&red_abs, pabs);
  __syncthreads();
  if (tid == 0) {
    out[OFF_L2] = (red_sum - red_trace) * (1.f/4096.f);
    out[OFF_NM] = red_trace * (1.f/4096.f);
    out[OFF_SP] = red_abs * (1.f/4096.f);
    out[OFF_L1] = 0.f;   // zero the L_sparse_1 accumulator every launch
  }

  // Gauss-Jordan (gram is SPD, well conditioned; no pivoting)
  for (int p = 0; p < 64; ++p) {
    if (tid < 64) colp[tid] = aug[tid][p];
    __syncthreads();
    float pivinv = 1.f / colp[p];
    if (tid < 128) aug[p][tid] *= pivinv;
    __syncthreads();
    for (int e = tid; e < 64*128; e += 1024) {
      int r = e >> 7, c = e & 127;
      if (r != p) aug[r][c] -= colp[r] * aug[p][c];
    }
    __syncthreads();
  }

  // U = C^T @ W^T  [64][10]
  for (int e = tid; e < 64*CCLS; e += 1024) {
    int j = e / CCLS, cls = e % CCLS;
    float acc = 0.f;
    for (int d = 0; d < D_DIM; ++d) acc += Cc[d*NCON + j] * W[cls*D_DIM + d];
    sU[j][cls] = acc;
  }
  __syncthreads();
  // Vmat = Ginv @ U -> ws[0..639]
  for (int e = tid; e < 64*CCLS; e += 1024) {
    int j = e / CCLS, cls = e % CCLS;
    float acc = 0.f;
    for (int t = 0; t < 64; ++t) acc += aug[j][64 + t] * sU[t][cls];
    ws[j*CCLS + cls] = acc;
  }
}

// ---------------------------------------------------------------------------
// Kernel 2: t_norm2[n] = sum_d T[d,n]^2  (coalesced column sums)
// ---------------------------------------------------------------------------
__global__ void k_tnorm(const float* __restrict__ T, float* __restrict__ tn2) {
  int n = blockIdx.x * blockDim.x + threadIdx.x;
  if (n >= N_TRAIN) return;
  float acc = 0.f;
  for (int d = 0; d < D_DIM; ++d) {
    float v = T[(size_t)d*N_TRAIN + n];
    acc += v * v;
  }
  tn2[n] = acc;
}

// ---------------------------------------------------------------------------
// Kernel 3: CT = concept^T @ T  -> [64][N], f32 WMMA 16x16x4 with
// double-buffered async global->LDS staging of the streamed B operand.
//
// Block = 256 threads = 8 waves, computes all 64 concept rows x 32 train cols.
// K loop in slabs of 32: each slab = T[k0:k0+32, n0:n0+32] = 4KB, staged via
// one GLOBAL_LOAD_ASYNC_TO_LDS_B128 per thread (ASYNCcnt, in-order), while
// WMMA consumes the previous slab from LDS.
// ---------------------------------------------------------------------------
#define KSLAB 32
#define NSLAB (D_DIM / KSLAB)   // 24

__global__ __launch_bounds__(256) void k_ct_gemm(const float* __restrict__ Cc,  // [D][64]
                                                 const float* __restrict__ T,   // [D][N]
                                                 float* __restrict__ CT) {
  __shared__ __align__(16) float Tbuf[2][KSLAB][32];   // 8KB double buffer

  int tid  = threadIdx.x;
  int wave = tid >> 5;
  int lane = tid & 31;
  int m0 = (wave & 3) * 16;              // concept-row tile
  int nsub = wave >> 2;                  // 0..1 column sub-tile
  int n0blk = blockIdx.x * 32;
  int m  = lane & 15;
  int kh = lane >> 4;                    // lanes 16-31 carry K-pair {2,3}
  int nn = lane & 15;

  // per-thread async-copy assignment: row r (0..31), 4-float column group
  int cp_r  = tid >> 3;                  // 32 rows
  int cp_c4 = (tid & 7) * 4;             // 8 groups of 4 floats = 32 cols
  const float* gsrc_base = T + (size_t)cp_r * N_TRAIN + n0blk + cp_c4;
  unsigned lds0 = (unsigned)(uintptr_t)&Tbuf[0][cp_r][cp_c4];
  unsigned lds1 = (unsigned)(uintptr_t)&Tbuf[1][cp_r][cp_c4];

  // prologue: issue slab 0 into buffer 0
  {
    const float* g = gsrc_base;          // k0 = 0
    asm volatile("global_load_async_to_lds_b128 %0, %1, off"
                 :: "v"(lds0), "v"(g) : "memory");
  }

  const float* ca = Cc + m0 + m;         // A[m,k] = concept[k][m0+m]
  v8f acc = {};

  for (int s = 0; s < NSLAB; ++s) {
    int pb = s & 1;
    // make sure every wave finished reading the buffer we are about to refill
    __syncthreads();
    if (s + 1 < NSLAB) {
      const float* g = gsrc_base + (size_t)(s + 1) * KSLAB * N_TRAIN;
      unsigned dst = (pb ? lds0 : lds1);
      asm volatile("global_load_async_to_lds_b128 %0, %1, off"
                   :: "v"(dst), "v"(g) : "memory");
      // previous slab (this iteration's data) complete when <=1 outstanding
      asm volatile("s_wait_asynccnt 0x1" ::: "memory");
    } else {
      asm volatile("s_wait_asynccnt 0x0" ::: "memory");
    }
    __syncthreads();   // slab s visible to all waves

    int k0 = s * KSLAB;
    #pragma unroll
    for (int kk = 0; kk < KSLAB; kk += 4) {
      int ka = kk + 2*kh;
      v2f a, b;
      a.x = ca[(size_t)(k0 + ka) * NCON];
      a.y = ca[(size_t)(k0 + ka + 1) * NCON];
      b.x = Tbuf[pb][ka][nsub*16 + nn];
      b.y = Tbuf[pb][ka + 1][nsub*16 + nn];
      acc = __builtin_amdgcn_wmma_f32_16x16x4_f32(
          /*neg_a=*/false, a, /*neg_b=*/false, b,
          /*c_mod=*/(short)0, acc, /*reuse_a=*/false, /*reuse_b=*/false);
    }
  }

  // C layout: vgpr v, lanes 0-15 -> M=v, lanes 16-31 -> M=v+8; N = lane&15
  int col = lane & 15;
  int n0 = n0blk + nsub * 16;
  #pragma unroll
  for (int v = 0; v < 8; ++v) {
    CT[(size_t)(m0 + v + 8*kh) * N_TRAIN + n0 + col] = acc[v];
  }
}

// ---------------------------------------------------------------------------
// Kernel 4: per 32-row block of the batch:
//   A = E @ concept (WMMA, into LDS), then
//   orig_pred = E @ W^T + b ; y_pred = A @ Vmat + b
// ---------------------------------------------------------------------------
__global__ __launch_bounds__(256) void k_preds(const float* __restrict__ E,     // [B][D]
                                               const float* __restrict__ Cc,    // [D][64]
                                               const float* __restrict__ W,     // [10][D]
                                               const float* __restrict__ bias,  // [10]
                                               const float* __restrict__ Vmat,  // [64][10]
                                               float* __restrict__ out) {
  __shared__ float sA[32][NCON + 1];
  int wave = threadIdx.x >> 5;
  int lane = threadIdx.x & 31;
  int rt = wave >> 2;                   // 0..1 row tile
  int c0 = (wave & 3) * 16;             // concept-column tile
  int r0g = blockIdx.x * 32 + rt * 16;
  int m  = lane & 15;
  int kh = lane >> 4;
  v8f acc = {};
  const float* ea = E + (size_t)(r0g + m) * D_DIM;
  const float* cb = Cc + c0 + (lane & 15);
  #pragma unroll 4
  for (int k = 0; k < D_DIM; k += 4) {
    int ka = k + 2*kh;
    v2f a, b;
    a.x = ea[ka];
    a.y = ea[ka + 1];
    b.x = cb[(size_t)ka * NCON];
    b.y = cb[(size_t)(ka+1) * NCON];
    acc = __builtin_amdgcn_wmma_f32_16x16x4_f32(false, a, false, b, (short)0, acc, false, false);
  }
  int col = lane & 15;
  #pragma unroll
  for (int v = 0; v < 8; ++v) sA[rt*16 + v + 8*kh][c0 + col] = acc[v];
  __syncthreads();

  int gr0 = blockIdx.x * 32;
  for (int p = threadIdx.x; p < 32 * CCLS; p += 256) {
    int row = p / CCLS, cls = p % CCLS;
    const float* er = E + (size_t)(gr0 + row) * D_DIM;
    const float* wr = W + cls * D_DIM;
    float o = 0.f;
    for (int d = 0; d < D_DIM; ++d) o += er[d] * wr[d];
    float y = 0.f;
    for (int j = 0; j < NCON; ++j) y += sA[row][j] * Vmat[j*CCLS + cls];
    out[OFF_ORIG  + (size_t)(gr0 + row)*CCLS + cls] = o + bias[cls];
    out[OFF_YPRED + (size_t)(gr0 + row)*CCLS + cls] = y + bias[cls];
  }
}

// ---------------------------------------------------------------------------
// Kernel 5: per-concept exact top-k(sum) via histogram selection.
// key s_j = 2*CT[c,j] - tn2[j]  (k largest s == k smallest distances)
// L_sparse_1 += (sum of CT over selected)/k / NCON  via global f32 atomic.
// ---------------------------------------------------------------------------
__global__ __launch_bounds__(1024) void k_topk(const float* __restrict__ CT,
                                               const float* __restrict__ tn2,
                                               const int* __restrict__ topk,
                                               float* __restrict__ out) {
  __shared__ float sred[1024];
  __shared__ unsigned hist[1024];
  __shared__ float cand_s[1024];
  __shared__ float cand_ct[1024];
  __shared__ float s_lo, s_hi, sum_high, sum_sel;
  __shared__ int bin_t, need_s, cand_cnt;
  int c = blockIdx.x;
  int tid = threadIdx.x;
  int k = topk[0];
  const float* row = CT + (size_t)c * N_TRAIN;

  // pass 1: min/max of key
  float lmax = -3.4e38f, lmin = 3.4e38f;
  for (int j = tid; j < N_TRAIN; j += 1024) {
    float s = 2.f*row[j] - tn2[j];
    lmax = fmaxf(lmax, s);
    lmin = fminf(lmin, s);
  }
  sred[tid] = lmax; __syncthreads();
  for (int o = 512; o > 0; o >>= 1) { if (tid < o) sred[tid] = fmaxf(sred[tid], sred[tid+o]); __syncthreads(); }
  if (tid == 0) s_hi = sred[0];
  __syncthreads();
  sred[tid] = lmin; __syncthreads();
  for (int o = 512; o > 0; o >>= 1) { if (tid < o) sred[tid] = fminf(sred[tid], sred[tid+o]); __syncthreads(); }
  if (tid == 0) s_lo = sred[0];
  hist[tid] = 0u;
  if (tid == 0) { sum_high = 0.f; sum_sel = 0.f; cand_cnt = 0; }
  __syncthreads();

  float lo = s_lo, width = s_hi - s_lo;
  if (width <= 0.f) {               // degenerate: all keys equal
    if (tid == 0) {
      float s = 0.f;
      for (int j = 0; j < k; ++j) s += row[j];
      atomicAdd(&out[OFF_L1], s / (float)k * (1.f/(float)NCON));
    }
    return;
  }
  float scale = 1024.f / width;

  // pass 2: histogram (ds_add atomics)
  for (int j = tid; j < N_TRAIN; j += 1024) {
    float s = 2.f*row[j] - tn2[j];
    int b = (int)((s - lo) * scale);
    b = b < 0 ? 0 : (b > 1023 ? 1023 : b);
    atomicAdd(&hist[b], 1u);
  }
  __syncthreads();

  // suffix count from top bin: find boundary bin t
  if (tid == 0) {
    unsigned cum = 0; int t = 0; unsigned chi = 0;
    for (int b2 = 1023; b2 >= 0; --b2) {
      unsigned nc2 = cum + hist[b2];
      if (nc2 >= (unsigned)k) { t = b2; chi = cum; break; }
      cum = nc2;
    }
    bin_t = t; need_s = k - (int)chi;
  }
  __syncthreads();
  int t = bin_t;

  // pass 3: sum bins above t, collect boundary candidates
  for (int j = tid; j < N_TRAIN; j += 1024) {
    float ctv = row[j];
    float s = 2.f*ctv - tn2[j];
    int b = (int)((s - lo) * scale);
    b = b < 0 ? 0 : (b > 1023 ? 1023 : b);
    if (b > t) {
      atomicAdd(&sum_high, ctv);
    } else if (b == t) {
      int i = atomicAdd(&cand_cnt, 1);
      if (i < 1024) { cand_s[i] = s; cand_ct[i] = ctv; }
    }
  }
  __syncthreads();

  // rank-select the `need` largest boundary candidates
  int m2 = cand_cnt; if (m2 > 1024) m2 = 1024;
  int need = need_s; if (need > m2) need = m2; if (need < 0) need = 0;
  for (int i = tid; i < m2; i += 1024) {
    float si = cand_s[i];
    int rank = 0;
    for (int j = 0; j < m2; ++j) {
      float sj = cand_s[j];
      if (sj > si || (sj == si && j < i)) ++rank;
    }
    if (rank < need) atomicAdd(&sum_sel, cand_ct[i]);
  }
  __syncthreads();
  if (tid == 0) {
    atomicAdd(&out[OFF_L1], (sum_high + sum_sel) / (float)k * (1.f/(float)NCON));
  }
}

// ---------------------------------------------------------------------------
extern "C" void kernel_launch(void* const* d_in, const int* in_sizes, int n_in,
                              void* d_out, int out_size, void* d_ws, size_t ws_size,
                              hipStream_t stream) {
  const float* E    = (const float*)d_in[0];   // [8192][768]
  // d_in[1] = h_x, unused by the reference outputs
  const float* Cc   = (const float*)d_in[2];   // [768][64]
  const float* T    = (const float*)d_in[3];   // [768][100000]
  const float* W    = (const float*)d_in[4];   // [10][768]
  const float* bias = (const float*)d_in[5];   // [10]
  const int*  topk  = (const int*)d_in[6];     // scalar

  float* out  = (float*)d_out;
  float* ws   = (float*)d_ws;
  float* Vmat = ws;                 // 640 floats
  float* tn2  = ws + 1024;          // 100000 floats
  float* CT   = ws + 102400;        // 64*100000 floats (25.6 MB)

  k_setup <<<1, 1024, 0, stream>>>(Cc, W, out, ws);
  k_tnorm <<<(N_TRAIN + 255) / 256, 256, 0, stream>>>(T, tn2);
  k_ct_gemm<<<N_TRAIN / 32, 256, 0, stream>>>(Cc, T, CT);
  k_preds <<<BATCH / 32, 256, 0, stream>>>(E, Cc, W, bias, Vmat, out);
  k_topk  <<<NCON, 1024, 0, stream>>>(CT, tn2, topk, out);
}